// SpatialDeformableTransformer_63273458204877
// MI455X (gfx1250) — compile-verified
//
#include <hip/hip_runtime.h>
#include <hip/hip_bf16.h>

typedef __attribute__((ext_vector_type(2))) float v2f;
typedef __attribute__((ext_vector_type(8))) float v8f;

#define OUTR 512
#define INR  64
#define NB   8          // batches
#define NC   16         // channels of U
#define NFC  2          // flow channels
#define NCOL (INR*NFC)  // 128 fused (x,c) columns per flow row

// ---------------------------------------------------------------------------
// Kernel 0: normalized Keys-cubic (a=-0.5) resize weights, 64 -> 512.
// Dense W[512][64] for the WMMA GEMM + compact 4-tap table for the x-pass.
// Matches jax.image.resize(..., method="cubic") semantics: taps outside
// [0,63] dropped, remaining weights renormalized to sum to 1.
// ---------------------------------------------------------------------------
__global__ void build_weights_kernel(float* __restrict__ Wdense,
                                     float4* __restrict__ wtab,
                                     int* __restrict__ btab) {
    int o = blockIdx.x * blockDim.x + threadIdx.x;
    if (o >= OUTR) return;
    float s = (o + 0.5f) * (1.0f / 8.0f) - 0.5f;   // sample coord in input space
    int jf = (int)floorf(s);
    float w[4];
    float norm = 0.0f;
    #pragma unroll
    for (int k = 0; k < 4; ++k) {
        int j = jf - 1 + k;
        float d = fabsf(s - (float)j);
        float kv;
        if (d <= 1.0f)      kv = ((1.5f * d - 2.5f) * d) * d + 1.0f;
        else if (d < 2.0f)  kv = ((-0.5f * d + 2.5f) * d - 4.0f) * d + 2.0f;
        else                kv = 0.0f;
        if (j < 0 || j > INR - 1) kv = 0.0f;       // out-of-range tap dropped
        w[k] = kv;
        norm += kv;
    }
    float inv = 1.0f / norm;
    float* row = Wdense + o * INR;
    for (int j = 0; j < INR; ++j) row[j] = 0.0f;
    #pragma unroll
    for (int k = 0; k < 4; ++k) {
        w[k] *= inv;
        int j = jf - 1 + k;
        if (j >= 0 && j <= INR - 1) row[j] = w[k];
    }
    wtab[o] = make_float4(w[0], w[1], w[2], w[3]);
    btab[o] = jf - 1;
}

// ---------------------------------------------------------------------------
// Kernel 1: y-direction bicubic upsample as WMMA f32 GEMM.
//   temp1[b][oh][n] = sum_j W[oh][j] * flow[b][j][n],  n = x*2+c  (128 cols)
// One wave per 16x16 output tile; K=64 as 16 chained V_WMMA_F32_16X16X4_F32.
// ---------------------------------------------------------------------------
__global__ void upsample_y_wmma_kernel(const float* __restrict__ flow,
                                       const float* __restrict__ Wdense,
                                       float* __restrict__ temp1) {
    int wave = (blockIdx.x * blockDim.x + threadIdx.x) >> 5;  // 2048 waves
    int ln   = threadIdx.x & 31;
    int b  = wave >> 8;          // 256 tiles per batch
    int mt = (wave >> 3) & 31;   // 32 M-tiles (oh)
    int nt = wave & 7;           // 8  N-tiles (x*2+c)
    int col  = ln & 15;
    int half = ln >> 4;

    const float* F    = flow + (size_t)b * INR * NCOL;
    const float* Arow = Wdense + (mt * 16 + col) * INR;  // A row: M = col
    int nb = nt * 16 + col;                              // B/D column: N = col

    v8f acc = {};
    #pragma unroll
    for (int kb = 0; kb < 16; ++kb) {
        int k0 = kb * 4 + 2 * half;     // lanes 0-15: K0/K1, lanes 16-31: K2/K3
        v2f a;  a.x  = Arow[k0];              a.y  = Arow[k0 + 1];
        v2f bf; bf.x = F[k0 * NCOL + nb];     bf.y = F[(k0 + 1) * NCOL + nb];
        acc = __builtin_amdgcn_wmma_f32_16x16x4_f32(
            /*neg_a=*/false, a, /*neg_b=*/false, bf,
            /*c_mod=*/(short)0, acc, /*reuse_a=*/false, /*reuse_b=*/false);
    }
    float* outb = temp1 + (size_t)b * OUTR * NCOL;
    #pragma unroll
    for (int r = 0; r < 8; ++r)   // C/D: VGPR r -> M = r (+8 for upper half)
        outb[(mt * 16 + r + 8 * half) * NCOL + nb] = acc[r];
}

// ---------------------------------------------------------------------------
// Kernel 2: fused x-direction 4-tap bicubic + bilinear grid sample of U.
// 4 threads per output pixel, one float4 channel-quad each -> b128 loads/stores.
// ---------------------------------------------------------------------------
__global__ void deform_sample_kernel(const float* __restrict__ U,
                                     const float* __restrict__ temp1,
                                     const float4* __restrict__ wtab,
                                     const int* __restrict__ btab,
                                     float* __restrict__ out) {
    int tid = blockIdx.x * blockDim.x + threadIdx.x;
    int q   = tid & 3;          // channel quad (4 floats)
    int pix = tid >> 2;
    int ow = pix & (OUTR - 1);
    int oh = (pix >> 9) & (OUTR - 1);
    int b  = pix >> 18;

    // x-direction bicubic combine of the y-upsampled flow
    const float* t1 = temp1 + (size_t)(b * OUTR + oh) * NCOL;
    float4 wx = wtab[ow];
    int bx = btab[ow];
    float wk[4] = {wx.x, wx.y, wx.z, wx.w};
    float V0 = 0.0f, V1 = 0.0f;
    #pragma unroll
    for (int k = 0; k < 4; ++k) {
        int j = bx + k;
        j = j < 0 ? 0 : (j > INR - 1 ? INR - 1 : j);   // dropped taps have w=0
        V0 = fmaf(wk[k], t1[j * 2 + 0], V0);
        V1 = fmaf(wk[k], t1[j * 2 + 1], V1);
    }

    // identity grid + deformation, reference's (x+1)*W/2 mapping
    const float step = 2.0f / 511.0f;
    float xs = fmaf(step, (float)ow, -1.0f) + V0;
    float ys = fmaf(step, (float)oh, -1.0f) + V1;
    float x = (xs + 1.0f) * 256.0f;
    float y = (ys + 1.0f) * 256.0f;
    int x0 = (int)floorf(x), y0 = (int)floorf(y);
    int x1 = x0 + 1,         y1 = y0 + 1;
    x0 = min(max(x0, 0), 511); x1 = min(max(x1, 0), 511);
    y0 = min(max(y0, 0), 511); y1 = min(max(y1, 0), 511);
    float x0f = (float)x0, x1f = (float)x1, y0f = (float)y0, y1f = (float)y1;
    // weights from CLIPPED corners (faithful to reference)
    float wa = (x1f - x) * (y1f - y);
    float wb = (x1f - x) * (y - y0f);
    float wc = (x - x0f) * (y1f - y);
    float wd = (x - x0f) * (y - y0f);

    const float* Ub = U + (size_t)b * OUTR * OUTR * NC;
    float4 va = *((const float4*)(Ub + ((size_t)y0 * OUTR + x0) * NC) + q);
    float4 vb = *((const float4*)(Ub + ((size_t)y1 * OUTR + x0) * NC) + q);
    float4 vc = *((const float4*)(Ub + ((size_t)y0 * OUTR + x1) * NC) + q);
    float4 vd = *((const float4*)(Ub + ((size_t)y1 * OUTR + x1) * NC) + q);

    float4 r;
    r.x = wa * va.x + wb * vb.x + wc * vc.x + wd * vd.x;
    r.y = wa * va.y + wb * vb.y + wc * vc.y + wd * vd.y;
    r.z = wa * va.z + wb * vb.z + wc * vc.z + wd * vd.z;
    r.w = wa * va.w + wb * vb.w + wc * vc.w + wd * vd.w;

    *((float4*)(out + ((size_t)(b * OUTR + oh) * OUTR + ow) * NC) + q) = r;
}

// ---------------------------------------------------------------------------
extern "C" void kernel_launch(void* const* d_in, const int* in_sizes, int n_in,
                              void* d_out, int out_size, void* d_ws, size_t ws_size,
                              hipStream_t stream) {
    (void)in_sizes; (void)n_in; (void)out_size; (void)ws_size;
    const float* U    = (const float*)d_in[0];   // [8,512,512,16] f32
    const float* flow = (const float*)d_in[1];   // [8,64,64,2]    f32
    float* out = (float*)d_out;                  // [8,512,512,16] f32

    // workspace layout
    char* ws = (char*)d_ws;
    float*  Wdense = (float*)(ws);                           // 512*64*4  = 128 KB
    float4* wtab   = (float4*)(ws + 131072);                 // 512*16    =   8 KB
    int*    btab   = (int*)(ws + 131072 + 8192);             // 512*4     =   2 KB
    float*  temp1  = (float*)(ws + 141312);                  // 8*512*128*4 = 2 MB

    // 0) cubic resize weight tables
    build_weights_kernel<<<2, 256, 0, stream>>>(Wdense, wtab, btab);

    // 1) y-upsample GEMM via v_wmma_f32_16x16x4_f32
    //    2048 wave-tiles (8 b * 32 mt * 8 nt) -> 256 blocks x 8 waves
    upsample_y_wmma_kernel<<<256, 256, 0, stream>>>(flow, Wdense, temp1);

    // 2) fused x-upsample + bilinear gather: 8*512*512 pixels * 4 quads
    deform_sample_kernel<<<32768, 256, 0, stream>>>(U, temp1, wtab, btab, out);
}